// GraphVertEdgeNet_78408922956200
// MI455X (gfx1250) — compile-verified
//
#include <hip/hip_runtime.h>

// GraphVertEdgeNet for MI455X (gfx1250, wave32).
// fp32 WMMA (v_wmma_f32_16x16x4_f32) for all GEMMs; fused e-layer kernel
// (GEMM + bias + ev broadcast + ReLU + column-sum + residual, in-place).
// Weights staged to LDS via the Tensor Data Mover (tensor_load_to_lds, 6-arg
// clang-23 form); B fragments hoisted into registers so the steady-state
// inner loop is one ds_load_b64 + one v_wmma per K-step.

typedef __attribute__((ext_vector_type(2))) float v2f_;
typedef __attribute__((ext_vector_type(8))) float v8f_;

#define B_  64
#define N_  64
#define VF_ 32

__device__ __forceinline__ v8f_ wmma_f32(v2f_ a, v2f_ b, v8f_ c) {
  // 8-arg form: (neg_a, A, neg_b, B, c_mod, C, reuse_a, reuse_b)
  return __builtin_amdgcn_wmma_f32_16x16x4_f32(false, a, false, b, (short)0, c,
                                               false, false);
}

#ifndef __has_builtin
#define __has_builtin(x) 0
#endif

#if __has_builtin(__builtin_amdgcn_tensor_load_to_lds) && \
    __has_builtin(__builtin_amdgcn_s_wait_tensorcnt)
#define TDM_AVAILABLE 1
typedef __attribute__((ext_vector_type(4))) unsigned int tdm_u32x4;
typedef __attribute__((ext_vector_type(8))) int tdm_i32x8;
typedef __attribute__((ext_vector_type(4))) int tdm_i32x4;

// 1-D TDM load: nelem f32 elements, contiguous global -> contiguous LDS.
// D# per CDNA5 ISA (count=1, type=2 "image",
// data_size=2 (4B), tensor_dim0=tile_dim0=nelem, tensor_dim1=tile_dim1=1).
// clang-23 6-arg builtin: (g0 u32x4, g1 i32x8, g2 i32x4, g3 i32x4,
//                          g4 i32x8, cpol i32).
__device__ __forceinline__ void tdm_load_f32(const float* gsrc, float* lds_dst,
                                             unsigned nelem) {
  unsigned long long ga = (unsigned long long)(uintptr_t)gsrc;
  tdm_u32x4 g0 = {0u, 0u, 0u, 0u};
  g0[0] = 1u;                                   // count=1, user descriptor
  g0[1] = (unsigned)(uintptr_t)lds_dst;         // lds_addr (bytes)
  g0[2] = (unsigned)ga;                         // global_addr[31:0]
  g0[3] = ((unsigned)(ga >> 32) & 0x01FFFFFFu)  // global_addr[56:32]
          | (2u << 30);                         // type=2
  tdm_i32x8 g1 = {0, 0, 0, 0, 0, 0, 0, 0};
  g1[0] = (int)(2u << 16);                      // data_size=2 -> 4 bytes
  g1[1] = (int)((nelem & 0xFFFFu) << 16);       // tensor_dim0[15:0]
  g1[2] = (int)((nelem >> 16) & 0xFFFFu)        // tensor_dim0[31:16]
          | (1 << 16);                          // tensor_dim1 = 1
  g1[3] = (int)((nelem & 0xFFFFu) << 16);       // tile_dim0 (fits 16 bits)
  g1[4] = 1;                                    // tile_dim1 = 1
  g1[5] = (int)nelem;                           // tensor_dim0_stride[31:0]
  tdm_i32x4 g2 = {0, 0, 0, 0};
  tdm_i32x4 g3 = {0, 0, 0, 0};
  tdm_i32x8 g4 = {0, 0, 0, 0, 0, 0, 0, 0};
  __builtin_amdgcn_tensor_load_to_lds(g0, g1, g2, g3, g4, 0);
}
#endif

// ---------------------------------------------------------------------------
// BN over batch for [64, 2048] (bn_flat), biased variance, eps 1e-5.
__global__ void k_bn_flat(const float* __restrict__ x, const float* __restrict__ g,
                          const float* __restrict__ bb, float* __restrict__ out) {
  int c = blockIdx.x * blockDim.x + threadIdx.x;  // 0..2047
  float s = 0.f, ss = 0.f;
  for (int r = 0; r < B_; ++r) { float v = x[r * 2048 + c]; s += v; ss += v * v; }
  float mu  = s * (1.f / 64.f);
  float var = ss * (1.f / 64.f) - mu * mu;
  float sc  = rsqrtf(var + 1e-5f) * g[c];
  float bo  = bb[c] - mu * sc;
  for (int r = 0; r < B_; ++r) out[r * 2048 + c] = x[r * 2048 + c] * sc + bo;
}

// ---------------------------------------------------------------------------
// Generic v-path GEMM: out[4096,128] = act(A1[:,0:K1]@W[0:K1] + A2@W[K1:K1+K2]
//                                          + bias (+ resid)).
// Block = 256 threads (8 waves); block owns a 16-row tile, wave owns 16 cols.
template <int RELU>
__global__ __launch_bounds__(256)
void k_gemm_v(const float* __restrict__ A1, int K1,
              const float* __restrict__ A2, int K2,
              const float* __restrict__ Wt, const float* __restrict__ bias,
              const float* __restrict__ resid, float* __restrict__ out) {
  const int m0   = blockIdx.x * 16;
  const int wave = threadIdx.x >> 5;
  const int lane = threadIdx.x & 31;
  const int half = lane >> 4;                 // 0/1
  const int c    = (wave << 4) + (lane & 15); // output column 0..127
  const int mf   = lane & 15;                 // A-fragment row
  const int kb   = half * 2;                  // K sub-offset for this half-wave

  v8f_ acc = {0.f, 0.f, 0.f, 0.f, 0.f, 0.f, 0.f, 0.f};

  const float* Aseg = A1; int Kseg = K1; int kofs = 0;
  for (int seg = 0; seg < 2; ++seg) {
    if (Aseg) {
      const float* arow = Aseg + (size_t)(m0 + mf) * Kseg;
      for (int k0 = 0; k0 < Kseg; k0 += 4) {
        v2f_ a = *(const v2f_*)(arow + k0 + kb);           // A: M=mf, K=k0+kb,+1
        const float* wp = Wt + (size_t)(kofs + k0 + kb) * 128 + c;
        v2f_ bf; bf.x = wp[0]; bf.y = wp[128];             // B: K=k0+kb,+1; N=c
        acc = wmma_f32(a, bf, acc);
      }
      kofs += Kseg;
    }
    Aseg = A2; Kseg = K2;
  }

  float bc = bias[c];
#pragma unroll
  for (int k = 0; k < 8; ++k) {
    int row = m0 + k + half * 8;                           // C/D: M=k+8*half
    float v = acc[k] + bc;
    if (resid) v += resid[(size_t)row * 128 + c];
    if (RELU)  v = v > 0.f ? v : 0.f;
    out[(size_t)row * 128 + c] = v;
  }
}

// ---------------------------------------------------------------------------
// Fused e-layer: for block (b, j0..j0+15), loop i=0..63:
//   act = relu(e_in[b,i,j,:]@W + eb + ev[b,j,:] + ev[b,i,:])
//   vsum[b,j,:] += act          (per_v_e, pre-residual per reference)
//   e_out[b,i,j,:] = act (+ e_in if RES)
// W staged in LDS once (TDM when available); B fragments hoisted to regs.
template <int KIN, int RES>
__global__ __launch_bounds__(256)
void k_elayer(const float* __restrict__ ein, float* __restrict__ eout,
              const float* __restrict__ Wt, const float* __restrict__ eb,
              const float* __restrict__ ev, float* __restrict__ vsum) {
  extern __shared__ float sm[];
  const int KINP = KIN + 2;                    // padded LDS row stride (even)
  float* Ws = sm;                              // KIN*128 floats
  float* As = sm + KIN * 128;                  // 16*KINP floats

  const int b    = blockIdx.y;
  const int j0   = blockIdx.x * 16;
  const int tid  = threadIdx.x;
  const int wave = tid >> 5, lane = tid & 31, half = lane >> 4;
  const int c    = (wave << 4) + (lane & 15);
  const int mf   = lane & 15;
  const int kb   = half * 2;

  // ---- stage weights (KIN x 128) into LDS ----
#if TDM_AVAILABLE
  if (tid < 32) {                              // wave 0 drives the TDM
    tdm_load_f32(Wt, Ws, (unsigned)(KIN * 128));
    __builtin_amdgcn_s_wait_tensorcnt(0);      // data in LDS before barrier
  }
#else
  {
    const float4* src = (const float4*)Wt;
    float4* dst = (float4*)Ws;
    for (int idx = tid; idx < KIN * 32; idx += 256) dst[idx] = src[idx];
  }
#endif
  __syncthreads();

  // ---- hoist B fragments (loop-invariant over i) into registers ----
  v2f_ bfrag[KIN / 4];
#pragma unroll
  for (int k0 = 0; k0 < KIN; k0 += 4) {
    const float* wp = Ws + (k0 + kb) * 128 + c;
    bfrag[k0 / 4].x = wp[0];
    bfrag[k0 / 4].y = wp[128];
  }

  float bc = eb[c];
  float evj[8];
#pragma unroll
  for (int k = 0; k < 8; ++k)
    evj[k] = ev[(size_t)(b * 64 + j0 + k + half * 8) * 128 + c];

  v8f_ sum = {0.f, 0.f, 0.f, 0.f, 0.f, 0.f, 0.f, 0.f};

  for (int i = 0; i < 64; ++i) {
    const float* erow = ein + ((size_t)(b * 64 + i) * 64 + j0) * KIN;
    for (int idx = tid; idx < 16 * KIN; idx += 256)      // stage A tile (padded)
      As[(idx / KIN) * KINP + (idx % KIN)] = erow[idx];
    __syncthreads();
    if (i + 1 < 64)                                      // -> global_prefetch_b8
      __builtin_prefetch(ein + ((size_t)(b * 64 + i + 1) * 64 + j0) * KIN, 0, 0);

    float evi = ev[(size_t)(b * 64 + i) * 128 + c];
    v8f_ acc = {0.f, 0.f, 0.f, 0.f, 0.f, 0.f, 0.f, 0.f};
#pragma unroll
    for (int k0 = 0; k0 < KIN; k0 += 4) {
      v2f_ a = *(const v2f_*)(As + mf * KINP + k0 + kb);
      acc = wmma_f32(a, bfrag[k0 / 4], acc);
    }

    float* orow = eout + ((size_t)(b * 64 + i) * 64 + j0) * 128;
#pragma unroll
    for (int k = 0; k < 8; ++k) {
      int m = k + half * 8;                    // j within tile
      float v = acc[k] + bc + evj[k] + evi;
      v = v > 0.f ? v : 0.f;                   // relu'd pre-residual value
      sum[k] += v;                             // per_v_e uses e_next (pre-res)
      if (RES) v += As[m * KINP + c];          // e = e_next + e   (KIN==128)
      orow[(size_t)m * 128 + c] = v;
    }
    __syncthreads();
  }
#pragma unroll
  for (int k = 0; k < 8; ++k)
    vsum[(size_t)(b * 64 + j0 + k + half * 8) * 128 + c] = sum[k];
}

// ---------------------------------------------------------------------------
// bn_inner: channels = N axis, stats over (B, D) = 8192 values, eps = 128.0.
__global__ void k_bn_inner(const float* __restrict__ x, const float* __restrict__ g,
                           const float* __restrict__ bb, float* __restrict__ out) {
  __shared__ float rs[256], rss[256];
  const int n = blockIdx.x, tid = threadIdx.x;
  float s = 0.f, ss = 0.f;
  for (int idx = tid; idx < 8192; idx += 256) {
    int bi = idx >> 7, d = idx & 127;
    float v = x[((size_t)(bi * 64 + n) << 7) + d];
    s += v; ss += v * v;
  }
  rs[tid] = s; rss[tid] = ss; __syncthreads();
  for (int st = 128; st > 0; st >>= 1) {
    if (tid < st) { rs[tid] += rs[tid + st]; rss[tid] += rss[tid + st]; }
    __syncthreads();
  }
  float mu  = rs[0] * (1.f / 8192.f);
  float var = rss[0] * (1.f / 8192.f) - mu * mu;
  float sc  = rsqrtf(var + 128.0f) * g[n];
  float bo  = bb[n] - mu * sc;
  for (int idx = tid; idx < 8192; idx += 256) {
    int bi = idx >> 7, d = idx & 127;
    size_t o = ((size_t)(bi * 64 + n) << 7) + d;
    out[o] = x[o] * sc + bo;
  }
}

// --------------------------- layer-3 (width 1) -----------------------------
__global__ void k_ev3(const float* __restrict__ v, const float* __restrict__ w,
                      const float* __restrict__ b1, float* __restrict__ ev3) {
  int r = blockIdx.x * blockDim.x + threadIdx.x;  // 4096
  const float* row = v + (size_t)r * 128;
  float s = 0.f;
  for (int d = 0; d < 128; ++d) s += row[d] * w[d];
  ev3[r] = s + b1[0];
}

__global__ void k_e3(const float* __restrict__ e, const float* __restrict__ w,
                     const float* __restrict__ bb, const float* __restrict__ ev3,
                     float* __restrict__ oute) {
  int r = blockIdx.x * blockDim.x + threadIdx.x;  // 262144
  int b = r >> 12, i = (r >> 6) & 63, j = r & 63;
  const float* row = e + (size_t)r * 128;
  float s = 0.f;
  for (int d = 0; d < 128; ++d) s += row[d] * w[d];
  oute[r] = s + bb[0] + ev3[b * 64 + j] + ev3[b * 64 + i];  // no relu
}

__global__ void k_sum3(const float* __restrict__ oute, float* __restrict__ vsum3) {
  int r = blockIdx.x * blockDim.x + threadIdx.x;  // 4096 = b*64+j
  int b = r >> 6, j = r & 63;
  float s = 0.f;
  for (int i = 0; i < 64; ++i) s += oute[((size_t)(b * 64 + i) << 6) + j];
  vsum3[r] = s;
}

__global__ void k_v3(const float* __restrict__ vsum3, const float* __restrict__ v,
                     const float* __restrict__ w, const float* __restrict__ bb,
                     float* __restrict__ outv) {
  int r = blockIdx.x * blockDim.x + threadIdx.x;  // 4096
  const float* row = v + (size_t)r * 128;
  float s = vsum3[r] * w[0];                      // concat order: [per_v_e, v]
  for (int d = 0; d < 128; ++d) s += row[d] * w[1 + d];
  outv[r] = s + bb[0];                            // no relu
}

// ---------------------------------------------------------------------------
extern "C" void kernel_launch(void* const* d_in, const int* in_sizes, int n_in,
                              void* d_out, int out_size, void* d_ws, size_t ws_size,
                              hipStream_t stream) {
  const float* v_in    = (const float*)d_in[0];
  const float* e_in    = (const float*)d_in[1];
  const float* bn_in_g = (const float*)d_in[2];
  const float* bn_in_b = (const float*)d_in[3];
  const float* g1_evW  = (const float*)d_in[4];
  const float* g1_evb  = (const float*)d_in[5];
  const float* g1_eW   = (const float*)d_in[6];
  const float* g1_eb   = (const float*)d_in[7];
  const float* g1_vW   = (const float*)d_in[8];
  const float* g1_vb   = (const float*)d_in[9];
  const float* inn_evW = (const float*)d_in[10];
  const float* inn_evb = (const float*)d_in[11];
  const float* inn_eW  = (const float*)d_in[12];
  const float* inn_eb  = (const float*)d_in[13];
  const float* inn_vW  = (const float*)d_in[14];
  const float* inn_vb  = (const float*)d_in[15];
  const float* bn_g    = (const float*)d_in[16];
  const float* bn_b    = (const float*)d_in[17];
  const float* g3_evW  = (const float*)d_in[18];
  const float* g3_evb  = (const float*)d_in[19];
  const float* g3_eW   = (const float*)d_in[20];
  const float* g3_eb   = (const float*)d_in[21];
  const float* g3_vW   = (const float*)d_in[22];
  const float* g3_vb   = (const float*)d_in[23];

  float* W = (float*)d_ws;
  size_t off = 0;
  float* EBUF  = W + off; off += (size_t)33554432;  // e [64,64,64,128], 134 MB
  float* VBN   = W + off; off += 131072;            // bn'd input v [4096,32]
  float* EV    = W + off; off += 524288;            // ev [4096,128]
  float* VSUM  = W + off; off += 524288;            // per_v_e [4096,128]
  float* VCUR  = W + off; off += 524288;            // current v [4096,128]
  float* VTMP  = W + off; off += 524288;
  float* EV3   = W + off; off += 4096;
  float* VSUM3 = W + off; off += 4096;

  float* OUTV = (float*)d_out;        // [64,64,1]
  float* OUTE = OUTV + 4096;          // [64,64,64,1]

  k_bn_flat<<<dim3(8), dim3(256), 0, stream>>>(v_in, bn_in_g, bn_in_b, VBN);

  // ---- layer 1 ----
  k_gemm_v<0><<<dim3(256), dim3(256), 0, stream>>>(VBN, 32, nullptr, 0,
                                                   g1_evW, g1_evb, nullptr, EV);
  size_t sh1 = (size_t)(16 * 128 + 16 * (16 + 2)) * sizeof(float);
  k_elayer<16, 0><<<dim3(4, 64), dim3(256), sh1, stream>>>(e_in, EBUF, g1_eW,
                                                           g1_eb, EV, VSUM);
  k_gemm_v<1><<<dim3(256), dim3(256), 0, stream>>>(VSUM, 128, VBN, 32,
                                                   g1_vW, g1_vb, nullptr, VCUR);

  // ---- inner layers (residual + bn_inner) ----
  size_t sh = (size_t)(128 * 128 + 16 * (128 + 2)) * sizeof(float);
  for (int l = 0; l < 4; ++l) {
    k_gemm_v<0><<<dim3(256), dim3(256), 0, stream>>>(
        VCUR, 128, nullptr, 0, inn_evW + (size_t)l * 16384,
        inn_evb + l * 128, nullptr, EV);
    k_elayer<128, 1><<<dim3(4, 64), dim3(256), sh, stream>>>(
        EBUF, EBUF, inn_eW + (size_t)l * 16384, inn_eb + l * 128, EV, VSUM);
    k_gemm_v<1><<<dim3(256), dim3(256), 0, stream>>>(
        VSUM, 128, VCUR, 128, inn_vW + (size_t)l * 32768,
        inn_vb + l * 128, VCUR, VTMP);                    // resid fused
    k_bn_inner<<<dim3(64), dim3(256), 0, stream>>>(VTMP, bn_g + l * 64,
                                                   bn_b + l * 64, VCUR);
  }

  // ---- layer 3 (no relu, width-1 outputs) ----
  k_ev3<<<dim3(16), dim3(256), 0, stream>>>(VCUR, g3_evW, g3_evb, EV3);
  k_e3<<<dim3(1024), dim3(256), 0, stream>>>(EBUF, g3_eW, g3_eb, EV3, OUTE);
  k_sum3<<<dim3(16), dim3(256), 0, stream>>>(OUTE, VSUM3);
  k_v3<<<dim3(16), dim3(256), 0, stream>>>(VSUM3, VCUR, g3_vW, g3_vb, OUTV);
}